// RegressionLoss_45440753992375
// MI455X (gfx1250) — compile-verified
//
#include <hip/hip_runtime.h>

// SmoothL1 (beta=0.5) + per-element 1/len boundary mask + full sum.
// Bandwidth-bound: 192 MB traffic -> ~8.2 us floor at 23.3 TB/s.
// Two-pass deterministic reduction (no float atomics).

#define BETA        0.5f
#define HALF_BETA   0.25f

constexpr int      BLOCKS  = 4096;
constexpr int      THREADS = 256;                      // 8 wave32s per block
constexpr unsigned STRIDE  = (unsigned)BLOCKS * THREADS; // 1,048,576 — compile-time

__device__ __forceinline__ float sl1_masked(float o, float l, int len) {
    float d  = o - l;
    float ad = __builtin_fabsf(d);
    float sl1 = (ad < BETA) ? (d * d) : (ad - HALF_BETA);
    // mask: ad >= 1/len  <=>  ad * len >= 1   (len in [1,1000), exact in f32)
    return (ad * (float)len >= 1.0f) ? sl1 : 0.0f;
}

__global__ __launch_bounds__(THREADS)
void sl1_partial_kernel(const float* __restrict__ outs,
                        const float* __restrict__ labels,
                        const int*   __restrict__ lens,
                        float*       __restrict__ partial,
                        int n)
{
    const unsigned tid = blockIdx.x * THREADS + threadIdx.x;
    const unsigned n4  = (unsigned)n >> 2;

    const float4* __restrict__ o4  = reinterpret_cast<const float4*>(outs);
    const float4* __restrict__ l4  = reinterpret_cast<const float4*>(labels);
    const int4*   __restrict__ ln4 = reinterpret_cast<const int4*>(lens);

    float acc = 0.0f;

    // STRIDE is a power-of-two constant -> this is a single s_lshr, no division.
    const unsigned full = n4 / STRIDE;

    unsigned i = tid;
    for (unsigned it = 0; it < full; ++it, i += STRIDE) {
        // Scalar-uniform prefetch condition: one s_cbranch, then
        // global_prefetch_b8 for the next stride tile.
        if (it + 1 < full) {
            __builtin_prefetch(o4  + (i + STRIDE), 0, 1);
            __builtin_prefetch(l4  + (i + STRIDE), 0, 1);
            __builtin_prefetch(ln4 + (i + STRIDE), 0, 1);
        }
        float4 o  = o4[i];
        float4 l  = l4[i];
        int4   ln = ln4[i];
        acc += sl1_masked(o.x, l.x, ln.x);
        acc += sl1_masked(o.y, l.y, ln.y);
        acc += sl1_masked(o.z, l.z, ln.z);
        acc += sl1_masked(o.w, l.w, ln.w);
    }

    // Remainder float4 pass (only threads with i < n4 participate; at most one).
    if (i < n4) {
        float4 o  = o4[i];
        float4 l  = l4[i];
        int4   ln = ln4[i];
        acc += sl1_masked(o.x, l.x, ln.x);
        acc += sl1_masked(o.y, l.y, ln.y);
        acc += sl1_masked(o.z, l.z, ln.z);
        acc += sl1_masked(o.w, l.w, ln.w);
    }

    // Scalar tail for n not divisible by 4 (N here is; kept for generality).
    for (unsigned k = (n4 << 2) + tid; k < (unsigned)n; k += STRIDE) {
        acc += sl1_masked(outs[k], labels[k], lens[k]);
    }

    // Wave32 butterfly reduction (explicit width 32 — CDNA5 is wave32-only).
    #pragma unroll
    for (int off = 16; off > 0; off >>= 1)
        acc += __shfl_down(acc, off, 32);

    // Cross-wave reduction via LDS, fixed order -> deterministic.
    __shared__ float smem[THREADS / 32];
    if ((threadIdx.x & 31) == 0)
        smem[threadIdx.x >> 5] = acc;
    __syncthreads();

    if (threadIdx.x == 0) {
        float s = 0.0f;
        #pragma unroll
        for (int w = 0; w < THREADS / 32; ++w)
            s += smem[w];
        partial[blockIdx.x] = s;
    }
}

__global__ __launch_bounds__(256)
void sl1_final_kernel(const float* __restrict__ partial,
                      float*       __restrict__ out)
{
    // 4096 partials -> each thread folds 16 in fixed order.
    float v = 0.0f;
    #pragma unroll
    for (int j = 0; j < BLOCKS / 256; ++j)
        v += partial[threadIdx.x + j * 256];

    #pragma unroll
    for (int off = 16; off > 0; off >>= 1)
        v += __shfl_down(v, off, 32);

    __shared__ float smem[8];
    if ((threadIdx.x & 31) == 0)
        smem[threadIdx.x >> 5] = v;
    __syncthreads();

    if (threadIdx.x == 0) {
        float s = 0.0f;
        #pragma unroll
        for (int w = 0; w < 8; ++w)
            s += smem[w];
        out[0] = s;   // full overwrite every call; deterministic order throughout
    }
}

extern "C" void kernel_launch(void* const* d_in, const int* in_sizes, int n_in,
                              void* d_out, int out_size, void* d_ws, size_t ws_size,
                              hipStream_t stream)
{
    const float* outs    = reinterpret_cast<const float*>(d_in[0]);
    const float* labels  = reinterpret_cast<const float*>(d_in[1]);
    const int*   lens    = reinterpret_cast<const int*>(d_in[2]);
    float*       out     = reinterpret_cast<float*>(d_out);
    float*       partial = reinterpret_cast<float*>(d_ws);   // BLOCKS floats = 16 KB

    const int n = in_sizes[0];

    sl1_partial_kernel<<<BLOCKS, THREADS, 0, stream>>>(outs, labels, lens, partial, n);
    sl1_final_kernel<<<1, 256, 0, stream>>>(partial, out);
}